// GraphHook_78443282694767
// MI455X (gfx1250) — compile-verified
//
#include <hip/hip_runtime.h>
#include <hip/hip_bf16.h>
#include <math.h>

// ---------------- problem constants ----------------
#define BB    2
#define HH    192
#define WW    192
#define NN    (HH*WW)          // 36864 pixels
#define KSEG  256              // centroids
#define ITERS 5
#define KNN   5
#define TEMP  0.1f
// xy feature scale: (coord/H) * sqrt(COMPACT/100) = coord * sqrt(0.1)/192
#define XYSCALE (0.31622776601683794f/192.0f)

typedef float v2f __attribute__((ext_vector_type(2)));
typedef float v8f __attribute__((ext_vector_type(8)));

// ---------------- color conversion (matches reference) ----------------
__device__ __forceinline__ void rgb2lab(float r, float g, float b, float* lab) {
    float c[3] = { r, g, b };
    float lin[3];
#pragma unroll
    for (int i = 0; i < 3; ++i) {
        float v = fminf(fmaxf(c[i] * 0.5f + 0.5f, 0.0f), 1.0f);
        lin[i] = (v > 0.04045f) ? powf((v + 0.055f) * (1.0f/1.055f), 2.4f)
                                : v * (1.0f/12.92f);
    }
    float X = 0.412453f*lin[0] + 0.357580f*lin[1] + 0.180423f*lin[2];
    float Y = 0.212671f*lin[0] + 0.715160f*lin[1] + 0.072169f*lin[2];
    float Z = 0.019334f*lin[0] + 0.119193f*lin[1] + 0.950227f*lin[2];
    X *= (1.0f/0.95047f);
    Z *= (1.0f/1.08883f);
    float f0 = (X > 0.008856f) ? cbrtf(X) : 7.787f*X + (16.0f/116.0f);
    float f1 = (Y > 0.008856f) ? cbrtf(Y) : 7.787f*Y + (16.0f/116.0f);
    float f2 = (Z > 0.008856f) ? cbrtf(Z) : 7.787f*Z + (16.0f/116.0f);
    lab[0] = (116.0f*f1 - 16.0f) * 0.01f;   // L/100
    lab[1] = 500.0f*(f0 - f1);
    lab[2] = 200.0f*(f1 - f2);
}

// ---------------- pixel features: [B][N][8] = {Lab3, sy, sx, 0,0,0} --------
__global__ __launch_bounds__(256) void k_feat(const float* __restrict__ img,
                                              float* __restrict__ feat) {
    int idx = blockIdx.x * 256 + threadIdx.x;
    if (idx >= BB * NN) return;
    int b = idx / NN, p = idx % NN;
    int y = p / WW, x = p % WW;
    const float* ib = img + (size_t)b * 3 * NN;
    float lab[3];
    rgb2lab(ib[0*NN + p], ib[1*NN + p], ib[2*NN + p], lab);
    float* f = feat + (size_t)idx * 8;
    f[0] = lab[0]; f[1] = lab[1]; f[2] = lab[2];
    f[3] = (float)y * XYSCALE;
    f[4] = (float)x * XYSCALE;
    f[5] = 0.0f; f[6] = 0.0f; f[7] = 0.0f;
}

// ---------------- initial centroids on a 16x16 grid -----------------------
__global__ __launch_bounds__(256) void k_initcent(const float* __restrict__ feat,
                                                  float* __restrict__ cfeat,
                                                  float* __restrict__ ccc) {
    int idx = blockIdx.x * 256 + threadIdx.x;
    if (idx >= BB * KSEG) return;
    int b = idx / KSEG, k = idx % KSEG;
    int ky = k / 16, kx = k % 16;
    // linspace(0, 191, 16).astype(int32): float32 step then truncate
    int y = (int)((float)ky * (191.0f/15.0f));
    int x = (int)((float)kx * (191.0f/15.0f));
    int p = y * WW + x;
    const float* f = feat + ((size_t)b * NN + p) * 8;
    float* cf = cfeat + (size_t)idx * 8;
    float cc = 0.0f;
#pragma unroll
    for (int d = 0; d < 8; ++d) { cf[d] = f[d]; cc += f[d]*f[d]; }
    ccc[idx] = cc;
}

// ---------------- generic zero ----------------
__global__ __launch_bounds__(256) void k_zero(float* __restrict__ p, int n) {
    int i = blockIdx.x * 256 + threadIdx.x;
    if (i < n) p[i] = 0.0f;
}

// ---------------- assignment + accumulation (WMMA core) -------------------
// 1 wave per block; 8 sub-tiles of 16 pixels; K=256 centroids in 16 tiles.
// logits via V_WMMA_F32_16X16X4_F32 (two chained, K-slots 0..3 and 4..7).
__global__ __launch_bounds__(32) void k_assign(const float* __restrict__ feat,
                                               const float* __restrict__ cfeat,
                                               const float* __restrict__ ccc,
                                               const float* __restrict__ img,
                                               float* __restrict__ accum,
                                               int blocksPerBatch) {
    __shared__ float lw[16 * KSEG];   // logits -> softmax weights [pixel][cent]
    __shared__ float pay[16 * 8];     // payload per pixel: sy,sx,r,g,b

    const int b    = blockIdx.x / blocksPerBatch;
    const int grp  = blockIdx.x % blocksPerBatch;
    const int lane = threadIdx.x;     // 0..31
    const int half = lane >> 4;       // 0|1 selects K-slot pair / M half
    const int l16  = lane & 15;

    const float* featB = feat + (size_t)b * NN * 8;
    const float* cfB   = cfeat + (size_t)b * KSEG * 8;
    const float* ccB   = ccc + (size_t)b * KSEG;
    const float* imgB  = img + (size_t)b * 3 * NN;

    float acc[8][6];
#pragma unroll
    for (int j = 0; j < 8; ++j)
#pragma unroll
        for (int d = 0; d < 6; ++d) acc[j][d] = 0.0f;

    for (int sub = 0; sub < 8; ++sub) {
        const int pbase = grp * 128 + sub * 16;
        const int mypix = pbase + l16;
        const float* fp = featB + (size_t)mypix * 8;

        // A operands (16x4 f32 layout): lane<16 holds K0,K1; lane>=16 K2,K3
        v2f a_lo = { fp[half*2 + 0], fp[half*2 + 1] };
        v2f a_hi = { fp[half*2 + 4], fp[half*2 + 5] };

        if (lane < 16) {
            pay[l16*8 + 0] = fp[3];                 // sy
            pay[l16*8 + 1] = fp[4];                 // sx
            pay[l16*8 + 2] = imgB[0*NN + mypix];    // r
            pay[l16*8 + 3] = imgB[1*NN + mypix];    // g
            pay[l16*8 + 4] = imgB[2*NN + mypix];    // b
        }
        __syncthreads();

        // ---- pass A: logits for all 256 centroids via WMMA ----
        for (int t = 0; t < 16; ++t) {
            const int cent = t * 16 + l16;
            const float* cp = cfB + (size_t)cent * 8;
            v2f b_lo = { cp[half*2 + 0], cp[half*2 + 1] };
            v2f b_hi = { cp[half*2 + 4], cp[half*2 + 5] };
            v8f c8 = { 0.f, 0.f, 0.f, 0.f, 0.f, 0.f, 0.f, 0.f };
            c8 = __builtin_amdgcn_wmma_f32_16x16x4_f32(
                     false, a_lo, false, b_lo, (short)0, c8, false, false);
            c8 = __builtin_amdgcn_wmma_f32_16x16x4_f32(
                     false, a_hi, false, b_hi, (short)0, c8, false, false);
            float cc = ccB[cent];
#pragma unroll
            for (int r = 0; r < 8; ++r) {
                int m = r + half * 8;               // pixel row in tile
                // -dist/TEMP dropping per-pixel |p|^2 (cancels in softmax)
                float z = (2.0f * c8[r] - cc) * (1.0f / TEMP);
                lw[m * KSEG + cent] = z;
            }
        }
        __syncthreads();

        // ---- softmax over 256 centroids; lane pair (l16, l16+16) per pixel
        {
            const int m = l16;
            float mx = -3.0e38f;
            for (int j = 0; j < 128; ++j)
                mx = fmaxf(mx, lw[m * KSEG + half * 128 + j]);
            mx = fmaxf(mx, __shfl_xor(mx, 16, 32));
            float s = 0.0f;
            for (int j = 0; j < 128; ++j)
                s += __expf(lw[m * KSEG + half * 128 + j] - mx);
            s += __shfl_xor(s, 16, 32);
            float inv = 1.0f / s;
            for (int j = 0; j < 128; ++j) {
                int idx = m * KSEG + half * 128 + j;
                lw[idx] = __expf(lw[idx] - mx) * inv;
            }
        }
        __syncthreads();

        // ---- pass B: per-lane accumulation (8 centroids per lane) ----
#pragma unroll
        for (int j = 0; j < 8; ++j) {
            const int k = lane + 32 * j;
            for (int m = 0; m < 16; ++m) {
                float w = lw[m * KSEG + k];
                acc[j][0] += w;
                acc[j][1] += w * pay[m*8 + 0];
                acc[j][2] += w * pay[m*8 + 1];
                acc[j][3] += w * pay[m*8 + 2];
                acc[j][4] += w * pay[m*8 + 3];
                acc[j][5] += w * pay[m*8 + 4];
            }
        }
        __syncthreads();
    }

    // flush partials
    float* accB = accum + (size_t)b * KSEG * 8;
#pragma unroll
    for (int j = 0; j < 8; ++j) {
        const int k = lane + 32 * j;
#pragma unroll
        for (int d = 0; d < 6; ++d)
            atomicAdd(&accB[k*8 + d], acc[j][d]);
    }
}

// ---------------- centroid update: mean xy/rgb, Lab(mean rgb) -------------
__global__ __launch_bounds__(256) void k_update(const float* __restrict__ accum,
                                                float* __restrict__ cfeat,
                                                float* __restrict__ ccc,
                                                float* __restrict__ crgb) {
    int idx = blockIdx.x * 256 + threadIdx.x;
    if (idx >= BB * KSEG) return;
    const float* a = accum + (size_t)idx * 8;
    float sw  = a[0] + 1e-8f;
    float inv = 1.0f / sw;
    float sy = a[1]*inv, sx = a[2]*inv;
    float r  = a[3]*inv, g  = a[4]*inv, bl = a[5]*inv;
    float lab[3];
    rgb2lab(r, g, bl, lab);
    float* cf = cfeat + (size_t)idx * 8;
    cf[0] = lab[0]; cf[1] = lab[1]; cf[2] = lab[2];
    cf[3] = sy; cf[4] = sx; cf[5] = 0.0f; cf[6] = 0.0f; cf[7] = 0.0f;
    ccc[idx] = lab[0]*lab[0] + lab[1]*lab[1] + lab[2]*lab[2] + sy*sy + sx*sx;
    float* cr = crgb + (size_t)idx * 4;
    cr[0] = r; cr[1] = g; cr[2] = bl; cr[3] = 0.0f;
}

// ---------------- KNN(5) over centroid positions --------------------------
__global__ __launch_bounds__(256) void k_topk(const float* __restrict__ cfeat,
                                              int* __restrict__ topk) {
    const int b = blockIdx.x;
    const int k = threadIdx.x;
    __shared__ float cy[KSEG], cx[KSEG];
    cy[k] = cfeat[((size_t)b * KSEG + k) * 8 + 3];
    cx[k] = cfeat[((size_t)b * KSEG + k) * 8 + 4];
    __syncthreads();
    float my = cy[k], mx = cx[k];
    unsigned chosen[8] = {0,0,0,0,0,0,0,0};
    for (int r = 0; r < KNN; ++r) {
        float best = 3.0e38f; int bi = 0;
        for (int j = 0; j < KSEG; ++j) {
            if (j == k) continue;
            if ((chosen[j >> 5] >> (j & 31)) & 1u) continue;
            float dy = cy[j] - my, dx = cx[j] - mx;
            float d = dy*dy + dx*dx;
            if (d < best) { best = d; bi = j; }   // stable: lowest index on ties
        }
        chosen[bi >> 5] |= 1u << (bi & 31);
        topk[((size_t)b * KSEG + k) * 8 + r] = bi;
    }
}

// ---------------- Laplacian loss over KNN graph ---------------------------
__global__ __launch_bounds__(256) void k_loss(const int* __restrict__ topk,
                                              const float* __restrict__ crgb,
                                              float* __restrict__ scal) {
    int tid = blockIdx.x * 256 + threadIdx.x;   // over B*K*K
    float lsum = 0.0f, asum = 0.0f;
    if (tid < BB * KSEG * KSEG) {
        int b = tid / (KSEG * KSEG);
        int rem = tid % (KSEG * KSEG);
        int i = rem / KSEG, j = rem % KSEG;
        if (i != j) {
            const int* ti = topk + ((size_t)b * KSEG + i) * 8;
            const int* tj = topk + ((size_t)b * KSEG + j) * 8;
            bool adj = false;
#pragma unroll
            for (int r = 0; r < KNN; ++r)
                adj = adj || (ti[r] == j) || (tj[r] == i);
            if (adj) {
                const float* ci = crgb + ((size_t)b * KSEG + i) * 4;
                const float* cj = crgb + ((size_t)b * KSEG + j) * 4;
                float d0 = ci[0]-cj[0], d1 = ci[1]-cj[1], d2 = ci[2]-cj[2];
                lsum = d0*d0 + d1*d1 + d2*d2;
                asum = 1.0f;
            }
        }
    }
    __shared__ float sl[256], sa[256];
    sl[threadIdx.x] = lsum; sa[threadIdx.x] = asum;
    __syncthreads();
    for (int s = 128; s > 0; s >>= 1) {
        if (threadIdx.x < s) {
            sl[threadIdx.x] += sl[threadIdx.x + s];
            sa[threadIdx.x] += sa[threadIdx.x + s];
        }
        __syncthreads();
    }
    if (threadIdx.x == 0) {
        atomicAdd(&scal[0], sl[0]);
        atomicAdd(&scal[1], sa[0]);
    }
}

__global__ __launch_bounds__(32) void k_final(const float* __restrict__ scal,
                                              float* __restrict__ out) {
    if (threadIdx.x == 0 && blockIdx.x == 0) {
        float num_edges = scal[1] * 0.5f;
        out[0] = scal[0] / (num_edges + 1e-8f);
    }
}

// ---------------- host-side orchestration ----------------
extern "C" void kernel_launch(void* const* d_in, const int* in_sizes, int n_in,
                              void* d_out, int out_size, void* d_ws, size_t ws_size,
                              hipStream_t stream) {
    (void)in_sizes; (void)n_in; (void)out_size; (void)ws_size;
    const float* img = (const float*)d_in[0];
    float* out = (float*)d_out;
    float* ws  = (float*)d_ws;

    size_t off = 0;
    float* feat  = ws + off; off += (size_t)BB * NN * 8;      // pixel features
    float* cfeat = ws + off; off += (size_t)BB * KSEG * 8;    // centroid features
    float* ccc   = ws + off; off += (size_t)BB * KSEG;        // |c|^2
    float* crgb  = ws + off; off += (size_t)BB * KSEG * 4;    // centroid rgb
    float* accum = ws + off; off += (size_t)BB * KSEG * 8;    // 6 sums (+pad)
    int*   topk  = (int*)(ws + off); off += (size_t)BB * KSEG * 8;
    float* scal  = ws + off; off += 8;

    const int blocksPerBatch = NN / 128;   // 288: 128 pixels per block

    k_feat<<<(BB*NN + 255)/256, 256, 0, stream>>>(img, feat);
    k_initcent<<<(BB*KSEG + 255)/256, 256, 0, stream>>>(feat, cfeat, ccc);

    for (int it = 0; it < ITERS; ++it) {
        k_zero<<<(BB*KSEG*8 + 255)/256, 256, 0, stream>>>(accum, BB*KSEG*8);
        k_assign<<<BB * blocksPerBatch, 32, 0, stream>>>(feat, cfeat, ccc, img,
                                                         accum, blocksPerBatch);
        k_update<<<(BB*KSEG + 255)/256, 256, 0, stream>>>(accum, cfeat, ccc, crgb);
    }

    k_topk<<<BB, 256, 0, stream>>>(cfeat, topk);
    k_zero<<<1, 256, 0, stream>>>(scal, 8);
    k_loss<<<(BB*KSEG*KSEG + 255)/256, 256, 0, stream>>>(topk, crgb, scal);
    k_final<<<1, 32, 0, stream>>>(scal, out);
}